// LinearGrouping_37297495998973
// MI455X (gfx1250) — compile-verified
//
#include <hip/hip_runtime.h>
#include <math.h>

typedef float v2f __attribute__((ext_vector_type(2)));
typedef float v8f __attribute__((ext_vector_type(8)));

#define DIMD 128
#define NGRP 3

// ---------------------------------------------------------------------------
// Kernel 1: logits = nf @ W + b  via V_WMMA_F32_16X16X4_F32, then softmax.
// One wave32 handles a 16-node tile. A = 16x4 f32 tile of node features,
// B = 4x16 tile of W staged in LDS (zero-padded cols >= 3, transposed with
// stride 132 for bank-conflict-free ds_load_b64). Inner loop is branch-free.
// ---------------------------------------------------------------------------
__global__ void lg_coeff_wmma(const float* __restrict__ nf,
                              const float* __restrict__ W,
                              const float* __restrict__ b,
                              float* __restrict__ coeff,
                              int n_nodes) {
  __shared__ float Wt[16][132];             // Wt[col][k] = W[k][col], 0-padded
  __shared__ float lt[8][16][4];            // [wave][node-in-tile][group]

  // Stage W (128x3 -> 16x128 padded/transposed). 2048 entries, 256 threads.
  for (int i = threadIdx.x; i < 16 * DIMD; i += blockDim.x) {
    int col = i >> 7;                       // 0..15
    int row = i & (DIMD - 1);               // 0..127
    Wt[col][row] = (col < NGRP) ? W[row * NGRP + col] : 0.0f;
  }
  __syncthreads();

  const int lane = threadIdx.x & 31;
  const int wv   = threadIdx.x >> 5;
  const int tile = blockIdx.x * 8 + wv;
  const int node0 = tile * 16;
  const int half  = lane >> 4;              // 0: lanes 0-15, 1: lanes 16-31
  const int m     = lane & 15;

  // A fragment row (clamped for tail tiles; stores guarded later)
  int nodeA  = node0 + m;
  int nodeAc = nodeA < n_nodes ? nodeA : (n_nodes - 1);
  const float* arow = nf + (size_t)nodeAc * DIMD + half * 2;
  // B fragment row pointer in LDS (8-byte aligned for every k step)
  const float* brow = &Wt[m][half * 2];

  v8f c = {};
  for (int k = 0; k < DIMD; k += 4) {
    // A 16x4: VGPR0 holds K=k (lanes0-15) / K=k+2 (lanes16-31); VGPR1: +1
    v2f a;
    a.x = arow[k];
    a.y = arow[k + 1];
    // B 4x16: same K split, N striped across lanes (one ds_load_b64)
    float2 bv = *(const float2*)&brow[k];
    v2f bb;
    bb.x = bv.x;
    bb.y = bv.y;
    c = __builtin_amdgcn_wmma_f32_16x16x4_f32(
        /*neg_a=*/false, a, /*neg_b=*/false, bb,
        /*c_mod=*/(short)0, c, /*reuse_a=*/false, /*reuse_b=*/false);
  }

  // C/D layout: VGPR j, lanes 0-15 -> (M=j, N=lane); lanes 16-31 -> (M=j+8).
  if (m < NGRP) {
#pragma unroll
    for (int j = 0; j < 8; ++j)
      lt[wv][j + 8 * half][m] = c[j];
  }
  __syncthreads();

  if (lane < 16) {
    int node = node0 + lane;
    if (node < n_nodes) {
      float l0 = lt[wv][lane][0] + b[0];
      float l1 = lt[wv][lane][1] + b[1];
      float l2 = lt[wv][lane][2] + b[2];
      float mx = fmaxf(l0, fmaxf(l1, l2));
      float e0 = __expf(l0 - mx);
      float e1 = __expf(l1 - mx);
      float e2 = __expf(l2 - mx);
      float inv = 1.0f / (e0 + e1 + e2);
      coeff[node * NGRP + 0] = e0 * inv;
      coeff[node * NGRP + 1] = e1 * inv;
      coeff[node * NGRP + 2] = e2 * inv;
    }
  }
}

// ---------------------------------------------------------------------------
// Kernel 2: per-edge scatter. out[d,:] += dot(coeff[s],coeff[d]) * nf[s,:]
// One wave32 per edge: each lane covers 4 dims (global_load_b128 gather,
// 4 f32 atomics). nf (25.6MB) and out (25.6MB) are L2-resident on 192MB L2.
// ---------------------------------------------------------------------------
__global__ void lg_edge_scatter(const float* __restrict__ nf,
                                const int* __restrict__ src,
                                const int* __restrict__ dst,
                                const float* __restrict__ coeff,
                                float* __restrict__ acc,   // d_out accumulator
                                float* __restrict__ deg,
                                int n_edges) {
  const int lane = threadIdx.x & 31;
  const int wavesPerBlock = (int)(blockDim.x >> 5);
  int e = blockIdx.x * wavesPerBlock + (threadIdx.x >> 5);
  const int estride = gridDim.x * wavesPerBlock;
  for (; e < n_edges; e += estride) {
    int s = src[e];
    int d = dst[e];
    const float* cs = coeff + (size_t)s * NGRP;
    const float* cd = coeff + (size_t)d * NGRP;
    float w = cs[0] * cd[0] + cs[1] * cd[1] + cs[2] * cd[2];
    const float4 f = *(const float4*)(nf + (size_t)s * DIMD + lane * 4);
    float* dstp = acc + (size_t)d * DIMD + lane * 4;
    __hip_atomic_fetch_add(&dstp[0], f.x * w, __ATOMIC_RELAXED, __HIP_MEMORY_SCOPE_AGENT);
    __hip_atomic_fetch_add(&dstp[1], f.y * w, __ATOMIC_RELAXED, __HIP_MEMORY_SCOPE_AGENT);
    __hip_atomic_fetch_add(&dstp[2], f.z * w, __ATOMIC_RELAXED, __HIP_MEMORY_SCOPE_AGENT);
    __hip_atomic_fetch_add(&dstp[3], f.w * w, __ATOMIC_RELAXED, __HIP_MEMORY_SCOPE_AGENT);
    if (lane == 0)
      __hip_atomic_fetch_add(&deg[d], 1.0f, __ATOMIC_RELAXED, __HIP_MEMORY_SCOPE_AGENT);
  }
}

// ---------------------------------------------------------------------------
// Kernel 3: out[n,j] /= max(deg[n], 1)   (float4 per thread)
// ---------------------------------------------------------------------------
__global__ void lg_finalize(float* __restrict__ out,
                            const float* __restrict__ deg,
                            int n_nodes) {
  int idx = blockIdx.x * blockDim.x + threadIdx.x;   // one float4 per thread
  int total4 = n_nodes * (DIMD / 4);
  if (idx < total4) {
    float inv = 1.0f / fmaxf(deg[idx / (DIMD / 4)], 1.0f);
    float4* p = (float4*)out + idx;
    float4 v = *p;
    v.x *= inv; v.y *= inv; v.z *= inv; v.w *= inv;
    *p = v;
  }
}

extern "C" void kernel_launch(void* const* d_in, const int* in_sizes, int n_in,
                              void* d_out, int out_size, void* d_ws, size_t ws_size,
                              hipStream_t stream) {
  const float* nf  = (const float*)d_in[0];
  const int*   src = (const int*)d_in[1];
  const int*   dst = (const int*)d_in[2];
  const float* W   = (const float*)d_in[3];
  const float* b   = (const float*)d_in[4];
  float* out = (float*)d_out;

  const int n_nodes = in_sizes[0] / DIMD;
  const int n_edges = in_sizes[1];

  // Workspace layout: coeff [N*3], deg [N]  (~800 KB total)
  float* coeff = (float*)d_ws;
  float* deg   = coeff + (size_t)n_nodes * NGRP;

  // Deterministic init each call (harness poisons but never re-zeros).
  hipMemsetAsync(out, 0, (size_t)n_nodes * DIMD * sizeof(float), stream);
  hipMemsetAsync(deg, 0, (size_t)n_nodes * sizeof(float), stream);

  // 1) coefficients via WMMA (8 wave-tiles of 16 nodes per 256-thread block)
  int tiles   = (n_nodes + 15) / 16;
  int blocksA = (tiles + 7) / 8;
  lg_coeff_wmma<<<blocksA, 256, 0, stream>>>(nf, W, b, coeff, n_nodes);

  // 2) edge scatter: one wave per edge, grid-stride
  lg_edge_scatter<<<8192, 256, 0, stream>>>(nf, src, dst, coeff, out, deg, n_edges);

  // 3) mean normalization
  int total4 = n_nodes * (DIMD / 4);
  lg_finalize<<<(total4 + 255) / 256, 256, 0, stream>>>(out, deg, n_nodes);
}